// DWT2D_14104672600644
// MI455X (gfx1250) — compile-verified
//
#include <hip/hip_runtime.h>

// ---------------------------------------------------------------------------
// Single-level 2D Haar DWT, fused:  x(8,512,512,32) f32 -> out(8,256,256,128)
//
// Reference's two 512x512 GEMMs with the sparse Haar matrix collapse to a
// 2x2 butterfly per output position. Bandwidth-bound: 512 MiB total traffic
// @ 23.3 TB/s ~= 23 us. Strategy:
//   * one wave per 2 output positions; lanes 0-15 = channels (0..15),
//     lane>>4 selects input row (2p / 2p+1)
//   * 4 coalesced non-temporal b32 loads per position (2x64B segments each)
//   * butterfly executed by V_WMMA_F32_16X16X4_F32:
//       A(16x4)  = Haar rows {LL,LH,HL,HH} * 0.5 (replicated over M)
//       B(4x16)  = the 4 taps, columns = 16 channels
//       D[r][n]  = subband (r&3) of channel n   (lanes 0-15, VGPRs 0-3 useful)
//   * second channel-half results shuffled into lanes 16-31 (ds_bpermute) so
//     each subband store is a full-wave contiguous 128B non-temporal store
//   * every input byte read once, every output byte written once -> TH=NT
// ---------------------------------------------------------------------------

typedef float v2f __attribute__((ext_vector_type(2)));
typedef float v8f __attribute__((ext_vector_type(8)));

__global__ __launch_bounds__(256) void dwt2d_haar_wmma(
    const float* __restrict__ x, float* __restrict__ out)
{
    const int lane   = threadIdx.x & 31;
    const int waveId = (blockIdx.x << 3) | (threadIdx.x >> 5); // 8 waves/block
    const int pos0   = waveId << 1;                            // 2 positions/wave

    const int row = lane >> 4;   // 0 -> input row 2p, 1 -> 2p+1 (K=0,1 vs 2,3)
    const int c   = lane & 15;   // channel within 16-wide half

    // A(16x4) coefficients: A[i][k] = 0.5*H[i&3][k], i = lane&15 (so i&3).
    // H rows (LL,LH,HL,HH): k0 neg iff i0^i1; k1 neg iff i0; k2 neg iff i1; k3 pos.
    const int   i0 = lane & 1;
    const int   i1 = (lane >> 1) & 1;
    const float s0 = (row ? i1 : (i0 ^ i1)) ? -0.5f : 0.5f;  // k = row?2:0
    const float s1 = (row ? 0  : i0)        ? -0.5f : 0.5f;  // k = row?3:1
    const v2f amat  = {s0, s1};
    const v8f czero = {};

#pragma unroll
    for (int t = 0; t < 2; ++t) {
        const int pos = pos0 + t;          // pos = (b*256 + p)*256 + q
        const int b   = pos >> 16;
        const int p   = (pos >> 8) & 255;
        const int q   = pos & 255;

        // x index: ((b*512 + 2p+row)*512 + 2q)*32 + c
        const long ibase =
            ((((long)((b << 9) + 2 * p + row)) << 9) + (q << 1)) * 32 + c;

        // taps: +0 = (2q, ch c), +32 = (2q+1, ch c); +16/+48 = channels c+16
        const float t00 = __builtin_nontemporal_load(x + ibase + 0);
        const float t01 = __builtin_nontemporal_load(x + ibase + 32);
        const float t10 = __builtin_nontemporal_load(x + ibase + 16);
        const float t11 = __builtin_nontemporal_load(x + ibase + 48);
        const v2f b0 = {t00, t01};   // B(4x16): channels 0..15
        const v2f b1 = {t10, t11};   // B(4x16): channels 16..31

        // D = A * B  (EXEC is all-ones here: grid exactly tiles the work)
        v8f d0 = __builtin_amdgcn_wmma_f32_16x16x4_f32(
            false, amat, false, b0, (short)0, czero, false, false);
        v8f d1 = __builtin_amdgcn_wmma_f32_16x16x4_f32(
            false, amat, false, b1, (short)0, czero, false, false);

        // out base: pos*128; layout [LL(32)|LH(32)|HL(32)|HH(32)]
        float* op = out + ((long)pos << 7) + lane;
#pragma unroll
        for (int r = 0; r < 4; ++r) {
            // lanes 16-31 fetch channel-half-1 results from lanes 0-15 of d1
            const float hiv = __shfl(d1[r], c, 32);
            const float v   = (lane < 16) ? d0[r] : hiv;
            __builtin_nontemporal_store(v, op + (r << 5)); // 128B full-wave store
        }
    }
}

extern "C" void kernel_launch(void* const* d_in, const int* in_sizes, int n_in,
                              void* d_out, int out_size, void* d_ws, size_t ws_size,
                              hipStream_t stream) {
    // d_in[0] = x (float32, 8*512*512*32). d_in[1] = A — unused: the Haar
    // coefficients are baked into the WMMA A-matrix (A is a fixed constant
    // in the reference).
    const float* x = (const float*)d_in[0];
    float* out     = (float*)d_out;

    // 524288 output positions, 2 per wave, 8 waves per 256-thread block.
    dim3 grid(32768), block(256);
    hipLaunchKernelGGL(dwt2d_haar_wmma, grid, block, 0, stream, x, out);
}